// Net_skip_11390253269722
// MI455X (gfx1250) — compile-verified
//
#include <hip/hip_runtime.h>
#include <hip/hip_bf16.h>

// ---------------------------------------------------------------------------
// 3-layer GCN for MI455X (gfx1250, wave32).
// out[v] = dinv[v] * ( sum_{u->v} dinv[u]*h[u]  +  dinv[v]*h[v] ) + b
// Feature dim padded 30 -> 32 so one node row = 128B and one wave32 = one edge.
// Dense h = feat @ W done with V_WMMA_F32_16X16X4_F32 (full f32 WMMA).
// Edge kernel batches 32 edges/wave: coalesced index loads + v_readlane
// broadcast -> uniform (SGPR) node addresses for gather + atomic scatter.
// ---------------------------------------------------------------------------

typedef float v2f __attribute__((ext_vector_type(2)));
typedef float v8f __attribute__((ext_vector_type(8)));

// ---- small setup kernels ---------------------------------------------------

// Pad the three weight matrices into 32x32 row-major (K x Ncol) blocks.
__global__ void k_pad_weights(const float* __restrict__ W1,
                              const float* __restrict__ W2,
                              const float* __restrict__ W3,
                              float* __restrict__ Wp) {
  int j = threadIdx.x;            // 0..1023
  int r = j >> 5, c = j & 31;     // row (K), col (out feature)
  Wp[j]        = (r < 2 && c < 30) ? W1[r * 30 + c] : 0.0f;   // [2,30] -> 32x32
  Wp[1024 + j] = (c < 30)          ? W2[r * 30 + c] : 0.0f;   // [32,30]
  Wp[2048 + j] = (c == 0)          ? W3[r]          : 0.0f;   // [32,1]
}

__global__ void k_deg_init(int* __restrict__ deg, int N) {
  int t = blockIdx.x * blockDim.x + threadIdx.x;
  if (t < N) deg[t] = 1;          // self-loop
}

__global__ void k_edge_deg(const int* __restrict__ dst, int* __restrict__ deg, int E) {
  int t = blockIdx.x * blockDim.x + threadIdx.x;
  if (t < E) atomicAdd(&deg[dst[t]], 1);
}

__global__ void k_dinv(const int* __restrict__ deg, float* __restrict__ dinv, int N) {
  int t = blockIdx.x * blockDim.x + threadIdx.x;
  if (t < N) dinv[t] = rsqrtf((float)deg[t]);   // deg >= 1 always
}

// feat0[node, 0:2] = x[node], rest zero (padded K=32 layer-1 input)
__global__ void k_feat0(const float* __restrict__ x, float* __restrict__ feat, int N) {
  int t = blockIdx.x * blockDim.x + threadIdx.x;
  if (t < N * 32) {
    int node = t >> 5, c = t & 31;
    feat[t] = (c < 2) ? x[node * 2 + c] : 0.0f;
  }
}

// ---- dense: g = dinv .* (feat @ Wp), seed agg = g (self-loop term) ---------
// One wave per 16-node tile. 16x32 output via two 16x16 f32 WMMA chains, K=32.
__global__ void k_gemm_wmma(const float* __restrict__ feat,
                            const float* __restrict__ Wp,
                            const float* __restrict__ dinv,
                            float* __restrict__ g,
                            float* __restrict__ agg, int N) {
  int wid = (blockIdx.x * blockDim.x + threadIdx.x) >> 5;
  int ntiles = (N + 15) >> 4;
  if (wid >= ntiles) return;                 // whole-wave exit: EXEC stays all-1
  int lane = threadIdx.x & 31;
  int nloc = lane & 15;
  int half = lane >> 4;
  int m0   = wid << 4;

  // B fragments (4x16 f32): lane<16 holds K={0,1} (VGPR j -> K=2*half+j);
  // lane>=16 holds K={2,3}; N = lane&15.  Wp is 32(K) x 32(Ncol) row-major.
  v2f Bf0[8], Bf1[8];
#pragma unroll
  for (int kc = 0; kc < 8; ++kc) {
    int k0 = 4 * kc + 2 * half;
    Bf0[kc].x = Wp[(k0 + 0) * 32 + nloc];
    Bf0[kc].y = Wp[(k0 + 1) * 32 + nloc];
    Bf1[kc].x = Wp[(k0 + 0) * 32 + 16 + nloc];
    Bf1[kc].y = Wp[(k0 + 1) * 32 + 16 + nloc];
  }

  // A fragment (16x4 f32): lane holds M = lane&15, K pair = 2*half + {0,1}.
  int nodeA = m0 + nloc;
  if (nodeA >= N) nodeA = N - 1;             // clamp tail loads (stores guarded)
  const float* frow = feat + (size_t)nodeA * 32;

  v8f acc0 = {}; v8f acc1 = {};
#pragma unroll
  for (int kc = 0; kc < 8; ++kc) {
    v2f A;
    A.x = frow[4 * kc + 2 * half + 0];
    A.y = frow[4 * kc + 2 * half + 1];
    acc0 = __builtin_amdgcn_wmma_f32_16x16x4_f32(false, A, false, Bf0[kc],
                                                 (short)0, acc0, false, false);
    acc1 = __builtin_amdgcn_wmma_f32_16x16x4_f32(false, A, false, Bf1[kc],
                                                 (short)0, acc1, false, false);
  }

  // D layout: VGPR r -> row r (lanes 0-15) / row r+8 (lanes 16-31), col = lane&15.
  if (m0 + 16 <= N) {
    // full tile: branch-free epilogue (common case)
#pragma unroll
    for (int r = 0; r < 8; ++r) {
      int node = m0 + r + 8 * half;
      float dv = dinv[node];
      float v0 = dv * acc0[r];
      float v1 = dv * acc1[r];
      size_t base = (size_t)node * 32;
      g[base + nloc]        = v0;
      g[base + 16 + nloc]   = v1;
      agg[base + nloc]      = v0;   // seed with self-loop contribution
      agg[base + 16 + nloc] = v1;
    }
  } else {
#pragma unroll
    for (int r = 0; r < 8; ++r) {
      int node = m0 + r + 8 * half;
      if (node < N) {
        float dv = dinv[node];
        float v0 = dv * acc0[r];
        float v1 = dv * acc1[r];
        size_t base = (size_t)node * 32;
        g[base + nloc]        = v0;
        g[base + 16 + nloc]   = v1;
        agg[base + nloc]      = v0;
        agg[base + 16 + nloc] = v1;
      }
    }
  }
}

// ---- edge scatter: 32 edges per wave batch, lane = feature -----------------
// Coalesced 32-edge index load, then v_readlane broadcasts (s,d) into SGPRs
// so the gather/atomic use uniform node bases.  agg row = 128B line pair.
__global__ void k_edge_agg(const int* __restrict__ src, const int* __restrict__ dst,
                           const float* __restrict__ g, float* __restrict__ agg, int E) {
  int lane = threadIdx.x & 31;
  int wid  = (blockIdx.x * blockDim.x + threadIdx.x) >> 5;
  int nw   = (gridDim.x * blockDim.x) >> 5;

  for (int base = wid * 32; base < E; base += nw * 32) {
    int e  = base + lane;
    int ec = (e < E) ? e : (E - 1);          // clamp OOB lanes of tail batch
    int sv = src[ec];                        // coalesced 128B index loads
    int dv = dst[ec];
    int cnt = E - base; if (cnt > 32) cnt = 32;
    for (int i = 0; i < cnt; ++i) {
      int s = __builtin_amdgcn_readlane(sv, i);   // SGPR-uniform node ids
      int d = __builtin_amdgcn_readlane(dv, i);
      float v = g[s * 32 + lane];                 // coalesced 128B gather
      if (lane < 30) atomicAdd(&agg[d * 32 + lane], v);  // cols 30/31 are zero
    }
  }
}

// Layer 3 (single output feature): thread per edge.
__global__ void k_edge_agg1(const int* __restrict__ src, const int* __restrict__ dst,
                            const float* __restrict__ g, float* __restrict__ agg, int E) {
  int t = blockIdx.x * blockDim.x + threadIdx.x;
  if (t < E) atomicAdd(&agg[dst[t] * 32], g[src[t] * 32]);
}

// ---- finalize: feat_next = [ relu(dinv*agg + b) | x ]  (layers 1,2) --------
__global__ void k_finalize(const float* __restrict__ agg, const float* __restrict__ dinv,
                           const float* __restrict__ b, const float* __restrict__ x,
                           float* __restrict__ feat, int N) {
  int t = blockIdx.x * blockDim.x + threadIdx.x;
  if (t < N * 32) {
    int node = t >> 5, col = t & 31;
    float v;
    if (col < 30) {
      v = fmaxf(dinv[node] * agg[t] + b[col], 0.0f);
    } else {
      v = x[node * 2 + (col - 30)];       // concat original x
    }
    feat[t] = v;
  }
}

__global__ void k_out(const float* __restrict__ agg, const float* __restrict__ dinv,
                      const float* __restrict__ b3, float* __restrict__ out, int N) {
  int t = blockIdx.x * blockDim.x + threadIdx.x;
  if (t < N) out[t] = dinv[t] * agg[(size_t)t * 32] + b3[0];
}

// ---------------------------------------------------------------------------

extern "C" void kernel_launch(void* const* d_in, const int* in_sizes, int n_in,
                              void* d_out, int out_size, void* d_ws, size_t ws_size,
                              hipStream_t stream) {
  const float* x  = (const float*)d_in[0];   // [N,2]
  const int*   ei = (const int*)  d_in[1];   // [2,E] int
  const float* W1 = (const float*)d_in[2];
  const float* b1 = (const float*)d_in[3];
  const float* W2 = (const float*)d_in[4];
  const float* b2 = (const float*)d_in[5];
  const float* W3 = (const float*)d_in[6];
  const float* b3 = (const float*)d_in[7];
  float* out = (float*)d_out;

  int N = in_sizes[0] / 2;
  int E = in_sizes[1] / 2;
  const int* src = ei;
  const int* dst = ei + E;

  // workspace carve-out (256B aligned)
  char* ws = (char*)d_ws;
  size_t off = 0;
  auto take = [&](size_t bytes) -> char* {
    char* p = ws + off;
    off = (off + bytes + 255) & ~(size_t)255;
    return p;
  };
  int*   deg  = (int*)  take((size_t)N * 4);
  float* dinv = (float*)take((size_t)N * 4);
  float* feat = (float*)take((size_t)N * 32 * 4);
  float* g    = (float*)take((size_t)N * 32 * 4);
  float* agg  = (float*)take((size_t)N * 32 * 4);
  float* Wp   = (float*)take(3 * 1024 * 4);
  if (off > ws_size) return;  // insufficient scratch: leave d_out poisoned

  const int TB = 256;
  int gN   = (N + TB - 1) / TB;
  int gE   = (E + TB - 1) / TB;
  int gN32 = (N * 32 + TB - 1) / TB;
  int ntiles = (N + 15) / 16;
  int gTile  = (ntiles + 7) / 8;            // 8 waves per 256-thread block
  int gEdge  = 4096;                        // 32768 waves, grid-stride batches

  // --- setup (graph-invariant across layers) ---
  k_pad_weights<<<1, 1024, 0, stream>>>(W1, W2, W3, Wp);
  k_deg_init   <<<gN, TB, 0, stream>>>(deg, N);
  k_edge_deg   <<<gE, TB, 0, stream>>>(dst, deg, E);
  k_dinv       <<<gN, TB, 0, stream>>>(deg, dinv, N);
  k_feat0      <<<gN32, TB, 0, stream>>>(x, feat, N);

  // --- layer 1 ---
  k_gemm_wmma <<<gTile, TB, 0, stream>>>(feat, Wp + 0,    dinv, g, agg, N);
  k_edge_agg  <<<gEdge, TB, 0, stream>>>(src, dst, g, agg, E);
  k_finalize  <<<gN32, TB, 0, stream>>>(agg, dinv, b1, x, feat, N);

  // --- layer 2 ---
  k_gemm_wmma <<<gTile, TB, 0, stream>>>(feat, Wp + 1024, dinv, g, agg, N);
  k_edge_agg  <<<gEdge, TB, 0, stream>>>(src, dst, g, agg, E);
  k_finalize  <<<gN32, TB, 0, stream>>>(agg, dinv, b2, x, feat, N);

  // --- layer 3 (1 output feature) ---
  k_gemm_wmma <<<gTile, TB, 0, stream>>>(feat, Wp + 2048, dinv, g, agg, N);
  k_edge_agg1 <<<gE, TB, 0, stream>>>(src, dst, g, agg, E);
  k_out       <<<gN, TB, 0, stream>>>(agg, dinv, b3, out, N);
}